// HeteroSAGEEncoder_12352325943870
// MI455X (gfx1250) — compile-verified
//
#include <hip/hip_runtime.h>

#define NU 100000
#define NI 50000
#define NE 819200
#define FIN 128
#define HDIM 32

typedef __attribute__((ext_vector_type(2))) float v2f;
typedef __attribute__((ext_vector_type(8))) float v8f;

// ---------------------------------------------------------------------------
// Native CDNA5 fp32 global atomic add, no return (tracked by STOREcnt).
// Guarantees GLOBAL_ATOMIC_ADD_F32 instead of a possible CAS expansion.
// Device scope so the RMW is performed at the shared L2 atomic units.
// ---------------------------------------------------------------------------
__device__ __forceinline__ void atomic_add_f32_dev(float* p, float v) {
  asm volatile("global_atomic_add_f32 %0, %1, off scope:SCOPE_DEV"
               :
               : "v"(p), "v"(v)
               : "memory");
}

// ---------------------------------------------------------------------------
// GEMM: C[M,32] = X[M,K] @ W[32,K]^T  via V_WMMA_F32_16X16X4_F32
// One wave computes a 16-row x 32-col output tile (two 16x16 WMMA accums).
// M is always a multiple of 16 here (100000, 50000), so no row tails and
// EXEC is all-1s inside every wave that reaches the WMMA (ISA requirement).
//
// VGPR layouts (wave32, 32-bit):
//   A 16x4 : lane holds row M=lane%16; VGPR0/1 = K-pair {k0+2h, k0+2h+1}, h=lane/16
//   B 4x16 : lane holds col N=lane%16; same K-pair split  (B[k][n] = W[n][k])
//   C 16x16: VGPR r = row r (lanes 0-15) / row 8+r (lanes 16-31), col = lane%16
// ---------------------------------------------------------------------------
template <int K>
__global__ __launch_bounds__(256) void gemm_xwT_wmma(
    const float* __restrict__ X, const float* __restrict__ W,
    float* __restrict__ C, int M) {
  const int wave = blockIdx.x * (blockDim.x >> 5) + (threadIdx.x >> 5);
  const int lane = threadIdx.x & 31;
  const int row0 = wave * 16;
  if (row0 >= M) return;  // wave-uniform: whole wave exits together

  const int mr = lane & 15;   // row within tile (A) / col within tile (B)
  const int kh = lane >> 4;   // which K-pair of the 4-wide K step

  const float* __restrict__ xrow = X + (size_t)(row0 + mr) * K;
  const float* __restrict__ wn0  = W + (size_t)mr * K;         // cols 0..15
  const float* __restrict__ wn1  = W + (size_t)(mr + 16) * K;  // cols 16..31

  v8f acc0 = {};
  v8f acc1 = {};

#pragma unroll
  for (int k0 = 0; k0 < K; k0 += 4) {
    const int ka = k0 + 2 * kh;
    v2f a, b0, b1;
    a.x  = xrow[ka]; a.y  = xrow[ka + 1];
    b0.x = wn0[ka];  b0.y = wn0[ka + 1];
    b1.x = wn1[ka];  b1.y = wn1[ka + 1];
    acc0 = __builtin_amdgcn_wmma_f32_16x16x4_f32(false, a, false, b0,
                                                 (short)0, acc0, false, false);
    acc1 = __builtin_amdgcn_wmma_f32_16x16x4_f32(false, a, false, b1,
                                                 (short)0, acc1, false, false);
  }

  // store: VGPR r -> row (row0 + 8*kh + r), col (lane%16) [+16 for acc1]
  const int rbase = row0 + kh * 8;
#pragma unroll
  for (int r = 0; r < 8; ++r) {
    C[(size_t)(rbase + r) * HDIM + mr]      = acc0[r];
    C[(size_t)(rbase + r) * HDIM + 16 + mr] = acc1[r];
  }
}

// ---------------------------------------------------------------------------
// Degree count: one thread per edge. fp32 adds of 1.0 are exact (deg << 2^24).
// ---------------------------------------------------------------------------
__global__ __launch_bounds__(256) void count_deg_kernel(
    const int* __restrict__ dst, float* __restrict__ deg, int E) {
  int t = blockIdx.x * blockDim.x + threadIdx.x;
  if (t < E) atomic_add_f32_dev(&deg[dst[t]], 1.0f);
}

// ---------------------------------------------------------------------------
// Scatter-add transformed source features along edges.
// One wave per edge; lane = feature (coalesced 128B read + 128B atomic add).
// ---------------------------------------------------------------------------
__global__ __launch_bounds__(256) void scatter_feat_kernel(
    const float* __restrict__ feat, const int* __restrict__ src,
    const int* __restrict__ dst, float* __restrict__ agg, int E) {
  const int e = blockIdx.x * (blockDim.x >> 5) + (threadIdx.x >> 5);
  const int lane = threadIdx.x & 31;
  if (e >= E) return;
  const int s = src[e];
  const int d = dst[e];
  const float v = feat[(size_t)s * HDIM + lane];
  atomic_add_f32_dev(&agg[(size_t)d * HDIM + lane], v);
}

// ---------------------------------------------------------------------------
// SAGE epilogue: out = relu(l2norm(agg/max(deg,1) + bias + self))
// One wave per node, lane = feature; L2-norm via wave32 xor-shuffle reduce.
// ---------------------------------------------------------------------------
__global__ __launch_bounds__(256) void sage_finish_kernel(
    const float* __restrict__ agg, const float* __restrict__ deg,
    const float* __restrict__ selfterm, const float* __restrict__ bias,
    float* __restrict__ out, int N) {
  const int n = blockIdx.x * (blockDim.x >> 5) + (threadIdx.x >> 5);
  const int lane = threadIdx.x & 31;
  if (n >= N) return;
  const float d = fmaxf(deg[n], 1.0f);
  float v = agg[(size_t)n * HDIM + lane] / d + bias[lane] +
            selfterm[(size_t)n * HDIM + lane];
  float s = v * v;
#pragma unroll
  for (int off = 16; off > 0; off >>= 1) s += __shfl_xor(s, off, 32);
  v = v / fmaxf(sqrtf(s), 1e-12f);
  out[(size_t)n * HDIM + lane] = fmaxf(v, 0.0f);
}

// ---------------------------------------------------------------------------
// Projection epilogue (in place on d_out): io = l2norm(io + bias), no relu.
// ---------------------------------------------------------------------------
__global__ __launch_bounds__(256) void proj_finish_kernel(
    float* __restrict__ io, const float* __restrict__ bias, int N) {
  const int n = blockIdx.x * (blockDim.x >> 5) + (threadIdx.x >> 5);
  const int lane = threadIdx.x & 31;
  if (n >= N) return;
  float v = io[(size_t)n * HDIM + lane] + bias[lane];
  float s = v * v;
#pragma unroll
  for (int off = 16; off > 0; off >>= 1) s += __shfl_xor(s, off, 32);
  io[(size_t)n * HDIM + lane] = v / fmaxf(sqrtf(s), 1e-12f);
}

// ---------------------------------------------------------------------------

static inline int gemm_blocks(int M) { return ((M / 16) + 7) / 8; }  // 8 waves/blk
static inline int wave_blocks(int N) { return (N + 7) / 8; }

extern "C" void kernel_launch(void* const* d_in, const int* in_sizes, int n_in,
                              void* d_out, int out_size, void* d_ws, size_t ws_size,
                              hipStream_t stream) {
  const float* x_user    = (const float*)d_in[0];
  const float* x_item    = (const float*)d_in[1];
  const int*   edge_user = (const int*)d_in[2];
  const int*   edge_item = (const int*)d_in[3];
  const float* wl0_ui = (const float*)d_in[4];
  const float* b0_ui  = (const float*)d_in[5];
  const float* wr0_ui = (const float*)d_in[6];
  const float* wl0_iu = (const float*)d_in[7];
  const float* b0_iu  = (const float*)d_in[8];
  const float* wr0_iu = (const float*)d_in[9];
  const float* wl1_ui = (const float*)d_in[10];
  const float* b1_ui  = (const float*)d_in[11];
  const float* wr1_ui = (const float*)d_in[12];
  const float* wl1_iu = (const float*)d_in[13];
  const float* b1_iu  = (const float*)d_in[14];
  const float* wr1_iu = (const float*)d_in[15];
  const float* wp_user = (const float*)d_in[16];
  const float* bp_user = (const float*)d_in[17];
  const float* wp_item = (const float*)d_in[18];
  const float* bp_item = (const float*)d_in[19];

  float* out_user = (float*)d_out;                       // [NU,32]
  float* out_item = (float*)d_out + (size_t)NU * HDIM;   // [NI,32]

  // workspace layout (fp32)
  float* ws    = (float*)d_ws;
  float* deg_u = ws;                          // [NU]
  float* deg_i = deg_u + NU;                  // [NI]
  float* tl_u  = deg_i + NI;                  // [NU,32] msgs user->item
  float* tr_u  = tl_u + (size_t)NU * HDIM;    // [NU,32] user self term
  float* tl_i  = tr_u + (size_t)NU * HDIM;    // [NI,32] msgs item->user
  float* tr_i  = tl_i + (size_t)NI * HDIM;    // [NI,32] item self term
  float* agg_u = tr_i + (size_t)NI * HDIM;    // [NU,32]
  float* agg_i = agg_u + (size_t)NU * HDIM;   // [NI,32]
  float* user1 = agg_i + (size_t)NI * HDIM;   // [NU,32] (reused for user2)
  float* item1 = user1 + (size_t)NU * HDIM;   // [NI,32] (reused for item2)

  const dim3 B(256);

  // ---- degrees (same edge lists for both layers; compute once) ----
  hipMemsetAsync(deg_u, 0, (size_t)NU * sizeof(float), stream);
  hipMemsetAsync(deg_i, 0, (size_t)NI * sizeof(float), stream);
  count_deg_kernel<<<(NE + 255) / 256, B, 0, stream>>>(edge_user, deg_u, NE);
  count_deg_kernel<<<(NE + 255) / 256, B, 0, stream>>>(edge_item, deg_i, NE);

  // ---- layer 0: transform first (F=128 -> 32), then scatter 32 f/edge ----
  gemm_xwT_wmma<FIN><<<gemm_blocks(NU), B, 0, stream>>>(x_user, wl0_ui, tl_u, NU);
  gemm_xwT_wmma<FIN><<<gemm_blocks(NU), B, 0, stream>>>(x_user, wr0_iu, tr_u, NU);
  gemm_xwT_wmma<FIN><<<gemm_blocks(NI), B, 0, stream>>>(x_item, wl0_iu, tl_i, NI);
  gemm_xwT_wmma<FIN><<<gemm_blocks(NI), B, 0, stream>>>(x_item, wr0_ui, tr_i, NI);

  hipMemsetAsync(agg_i, 0, (size_t)NI * HDIM * sizeof(float), stream);
  hipMemsetAsync(agg_u, 0, (size_t)NU * HDIM * sizeof(float), stream);
  scatter_feat_kernel<<<wave_blocks(NE), B, 0, stream>>>(tl_u, edge_user, edge_item, agg_i, NE);
  scatter_feat_kernel<<<wave_blocks(NE), B, 0, stream>>>(tl_i, edge_item, edge_user, agg_u, NE);

  sage_finish_kernel<<<wave_blocks(NI), B, 0, stream>>>(agg_i, deg_i, tr_i, b0_ui, item1, NI);
  sage_finish_kernel<<<wave_blocks(NU), B, 0, stream>>>(agg_u, deg_u, tr_u, b0_iu, user1, NU);

  // ---- layer 1 (K=32) ----
  gemm_xwT_wmma<HDIM><<<gemm_blocks(NU), B, 0, stream>>>(user1, wl1_ui, tl_u, NU);
  gemm_xwT_wmma<HDIM><<<gemm_blocks(NU), B, 0, stream>>>(user1, wr1_iu, tr_u, NU);
  gemm_xwT_wmma<HDIM><<<gemm_blocks(NI), B, 0, stream>>>(item1, wl1_iu, tl_i, NI);
  gemm_xwT_wmma<HDIM><<<gemm_blocks(NI), B, 0, stream>>>(item1, wr1_ui, tr_i, NI);

  hipMemsetAsync(agg_i, 0, (size_t)NI * HDIM * sizeof(float), stream);
  hipMemsetAsync(agg_u, 0, (size_t)NU * HDIM * sizeof(float), stream);
  scatter_feat_kernel<<<wave_blocks(NE), B, 0, stream>>>(tl_u, edge_user, edge_item, agg_i, NE);
  scatter_feat_kernel<<<wave_blocks(NE), B, 0, stream>>>(tl_i, edge_item, edge_user, agg_u, NE);

  // user2/item2 overwrite user1/item1 (layer-1 GEMMs already consumed them)
  sage_finish_kernel<<<wave_blocks(NI), B, 0, stream>>>(agg_i, deg_i, tr_i, b1_ui, item1, NI);
  sage_finish_kernel<<<wave_blocks(NU), B, 0, stream>>>(agg_u, deg_u, tr_u, b1_iu, user1, NU);

  // ---- projections straight into d_out, then in-place bias + l2norm ----
  gemm_xwT_wmma<HDIM><<<gemm_blocks(NU), B, 0, stream>>>(user1, wp_user, out_user, NU);
  gemm_xwT_wmma<HDIM><<<gemm_blocks(NI), B, 0, stream>>>(item1, wp_item, out_item, NI);
  proj_finish_kernel<<<wave_blocks(NU), B, 0, stream>>>(out_user, bp_user, NU);
  proj_finish_kernel<<<wave_blocks(NI), B, 0, stream>>>(out_item, bp_item, NI);
}